// GAT_Conv_13649406067354
// MI455X (gfx1250) — compile-verified
//
#include <hip/hip_runtime.h>
#include <hip/hip_bf16.h>
#include <math.h>

typedef float v2f __attribute__((ext_vector_type(2)));
typedef float v8f __attribute__((ext_vector_type(8)));

#define NEG_SLOPE 0.2f

// ---------------------------------------------------------------- utilities
__global__ void fill_kernel(float* __restrict__ p, float v, int n) {
  int i = blockIdx.x * blockDim.x + threadIdx.x;
  if (i < n) p[i] = v;
}

__device__ __forceinline__ void atomicMaxF(float* addr, float val) {
  // monotone bit-pattern trick: non-negative floats compare as ints,
  // negative floats compare reversed as unsigned ints.
  if (val >= 0.f) atomicMax((int*)addr, __float_as_int(val));
  else            atomicMin((unsigned int*)addr, __float_as_uint(val));
}

// ---------------------------------------------------------------- WMMA GEMM
// H[M,Nout] = X[M,K] @ W[K,Nout], fp32 via V_WMMA_F32_16X16X4_F32.
// The whole W matrix (K*Nout*4 bytes <= 64KB here, WGP LDS is 320KB) is
// staged into LDS once per block with async global->LDS B128 copies
// (ASYNCcnt path), then each wave32 owns one 16-row M-tile and sweeps all
// 16-col N-tiles, feeding WMMA B operands from LDS (ds_load).
__global__ void wmma_gemm_lds(const float* __restrict__ X,
                              const float* __restrict__ W,
                              float* __restrict__ H,
                              int M, int K, int Nout) {
  extern __shared__ float smem[];          // K * Nout floats

  // ---- cooperative async stage of W into LDS (all threads, no divergence:
  // K*Nout is a multiple of 4*blockDim for all layers)
  const int total4 = (K * Nout) >> 2;      // float4 chunks
  for (int i = threadIdx.x; i < total4; i += blockDim.x) {
    unsigned lds = (unsigned)(uintptr_t)&smem[i * 4];  // low 32 bits = LDS addr
    const float* g = W + (size_t)i * 4;
    asm volatile("global_load_async_to_lds_b128 %0, %1, off"
                 :: "v"(lds), "v"(g) : "memory");
  }
  asm volatile("s_wait_asynccnt 0x0" ::: "memory");  // my async writes done
  __syncthreads();                                   // everyone's writes done

  const int lane  = threadIdx.x & 31;
  const int wave  = threadIdx.x >> 5;
  const int mtiles = (M + 15) >> 4;
  const int ntiles = Nout >> 4;
  const int mtile  = blockIdx.x * (blockDim.x >> 5) + wave;  // wave-uniform
  if (mtile >= mtiles) return;             // whole-wave skip, after barrier

  const int half = lane >> 4;              // 0 = lanes 0-15, 1 = lanes 16-31
  const int l16  = lane & 15;
  const int arow = mtile * 16 + l16;
  const int row  = arow < M ? arow : M - 1;  // clamp; stores guarded below
  const float* xrow = X + (size_t)row * K;

  for (int ntile = 0; ntile < ntiles; ++ntile) {
    const int col = ntile * 16 + l16;
    v8f acc = {};
    for (int k = 0; k < K; k += 4) {
      // A 16x4 fragment (ISA layout): lanes 0-15 -> K=k,k+1 ; 16-31 -> k+2,k+3
      const float* ap = xrow + k + 2 * half;
      v2f a; a.x = ap[0]; a.y = ap[1];
      // B 4x16 fragment from LDS: lanes 0-15 -> rows k,k+1 ; 16-31 -> k+2,k+3
      const float* bp = smem + (k + 2 * half) * Nout + col;
      v2f b; b.x = bp[0]; b.y = bp[Nout];
      acc = __builtin_amdgcn_wmma_f32_16x16x4_f32(
          /*neg_a=*/false, a, /*neg_b=*/false, b,
          /*c_mod=*/(short)0, acc, /*reuse_a=*/false, /*reuse_b=*/false);
    }
    // D layout: VGPR r -> row (r + 8*half), col = l16
#pragma unroll
    for (int r = 0; r < 8; ++r) {
      int m = mtile * 16 + r + 8 * half;
      if (m < M) H[(size_t)m * Nout + col] = acc[r];
    }
  }
}

// -------------------------------------------------- per-node attention dots
// s_src[i] = dot(h[i], a_src) ; s_dst[i] = dot(h[i], a_dst). One wave / node.
__global__ void node_dots(const float* __restrict__ H,
                          const float* __restrict__ a_src,
                          const float* __restrict__ a_dst,
                          float* __restrict__ s_src,
                          float* __restrict__ s_dst,
                          int n, int dim) {
  int node = (blockIdx.x * blockDim.x + threadIdx.x) >> 5;
  int lane = threadIdx.x & 31;
  if (node >= n) return;
  const float* h = H + (size_t)node * dim;
  float ps = 0.f, pd = 0.f;
  for (int d = lane; d < dim; d += 32) {
    float v = h[d];
    ps += v * a_src[d];
    pd += v * a_dst[d];
  }
  for (int off = 16; off; off >>= 1) {
    ps += __shfl_xor(ps, off, 32);
    pd += __shfl_xor(pd, off, 32);
  }
  if (lane == 0) { s_src[node] = ps; s_dst[node] = pd; }
}

// ------------------------------------------------------- edge-softmax chain
__global__ void edge_alpha_max(const int* __restrict__ eidx, int E, int EP,
                               const float* __restrict__ s_src,
                               const float* __restrict__ s_dst,
                               float* __restrict__ alpha,
                               float* __restrict__ amax) {
  int e = blockIdx.x * blockDim.x + threadIdx.x;
  if (e >= EP) return;
  int s = e < E ? eidx[e]     : e - E;   // self-loops appended
  int d = e < E ? eidx[E + e] : e - E;
  float a = s_src[s] + s_dst[d];
  a = a > 0.f ? a : NEG_SLOPE * a;       // leaky_relu
  alpha[e] = a;
  atomicMaxF(&amax[d], a);
}

__global__ void edge_exp_sum(const int* __restrict__ eidx, int E, int EP,
                             float* __restrict__ alpha,
                             const float* __restrict__ amax,
                             float* __restrict__ denom) {
  int e = blockIdx.x * blockDim.x + threadIdx.x;
  if (e >= EP) return;
  int d = e < E ? eidx[E + e] : e - E;
  float ex = __expf(alpha[e] - amax[d]);
  alpha[e] = ex;                          // reuse buffer for exp values
  atomicAdd(&denom[d], ex);
}

// One block per edge, `dim` lanes read one coalesced h[src] row and
// scatter-add the weighted message into out[dst].
__global__ void scatter_edges(const int* __restrict__ eidx, int E,
                              const float* __restrict__ H,
                              const float* __restrict__ ex,
                              const float* __restrict__ denom,
                              float* __restrict__ out, int dim) {
  int e = blockIdx.x;
  int t = threadIdx.x;
  int s = e < E ? eidx[e]     : e - E;
  int d = e < E ? eidx[E + e] : e - E;
  const float* hrow = H + (size_t)s * dim;
  __builtin_prefetch(hrow + t, 0, 3);     // global_prefetch_b8
  float w = ex[e] / (denom[d] + 1e-16f);
  atomicAdd(&out[(size_t)d * dim + t], hrow[t] * w);
}

__global__ void bias_elu(const float* __restrict__ acc,
                         const float* __restrict__ b,
                         float* __restrict__ xout, int total, int dim) {
  int i = blockIdx.x * blockDim.x + threadIdx.x;
  if (i >= total) return;
  float v = acc[i] + b[i % dim];
  xout[i] = v > 0.f ? v : expm1f(v);      // elu
}

// -------------------------------------------------------- final projection
// dim == 32 exactly: one lane per feature, wave reduction, sigmoid.
__global__ void final_linear_sigmoid(const float* __restrict__ X,
                                     const float* __restrict__ Wl,
                                     const float* __restrict__ bl,
                                     float* __restrict__ out, int n) {
  int node = (blockIdx.x * blockDim.x + threadIdx.x) >> 5;
  int lane = threadIdx.x & 31;
  if (node >= n) return;
  float p = X[(size_t)node * 32 + lane] * Wl[lane];
  for (int off = 16; off; off >>= 1) p += __shfl_xor(p, off, 32);
  if (lane == 0) out[node] = 1.f / (1.f + __expf(-(p + bl[0])));
}

// ---------------------------------------------------------------- launcher
extern "C" void kernel_launch(void* const* d_in, const int* in_sizes, int n_in,
                              void* d_out, int out_size, void* d_ws, size_t ws_size,
                              hipStream_t stream) {
  const float* X   = (const float*)d_in[0];
  const int*   EI  = (const int*)d_in[1];
  // d_in[2] = edge_weight: ignored (faithful to PyG GATConv with edge_dim=None)
  const float* W1  = (const float*)d_in[3];
  const float* as1 = (const float*)d_in[4];
  const float* ad1 = (const float*)d_in[5];
  const float* b1  = (const float*)d_in[6];
  const float* W2  = (const float*)d_in[7];
  const float* as2 = (const float*)d_in[8];
  const float* ad2 = (const float*)d_in[9];
  const float* b2  = (const float*)d_in[10];
  const float* W3  = (const float*)d_in[11];
  const float* as3 = (const float*)d_in[12];
  const float* ad3 = (const float*)d_in[13];
  const float* b3  = (const float*)d_in[14];
  const float* Wl  = (const float*)d_in[15];
  const float* bl  = (const float*)d_in[16];

  const int Nn = in_sizes[0] / 128;   // 50000
  const int E  = in_sizes[1] / 2;     // 800000
  const int EP = E + Nn;              // + self loops

  // workspace layout (floats)
  float* ws   = (float*)d_ws;
  size_t big  = (size_t)Nn * 128;
  float* H    = ws;                   // gemm output h       [N,128] max
  float* ACC  = H   + big;            // aggregated messages [N,128] max
  float* XB   = ACC + big;            // layer activations   [N,128] max
  float* SS   = XB  + big;            // s_src [N]
  float* SD   = SS  + Nn;             // s_dst [N]
  float* AMAX = SD  + Nn;             // segment max [N]
  float* DEN  = AMAX + Nn;            // segment sum [N]
  float* AEX  = DEN + Nn;             // per-edge alpha/exp [EP]

  auto layer = [&](const float* Xin, const float* W, const float* av,
                   const float* dv, const float* bv, int in, int outd,
                   float* xout) {
    const int mtiles = (Nn + 15) / 16;
    const size_t lds_bytes = (size_t)in * outd * sizeof(float);  // <= 64KB
    wmma_gemm_lds<<<(mtiles + 3) / 4, 128, lds_bytes, stream>>>(
        Xin, W, H, Nn, in, outd);
    node_dots<<<(Nn + 7) / 8, 256, 0, stream>>>(H, av, dv, SS, SD, Nn, outd);
    fill_kernel<<<(Nn + 255) / 256, 256, 0, stream>>>(AMAX, -INFINITY, Nn);
    fill_kernel<<<(Nn + 255) / 256, 256, 0, stream>>>(DEN, 0.f, Nn);
    const int total = Nn * outd;
    fill_kernel<<<(total + 255) / 256, 256, 0, stream>>>(ACC, 0.f, total);
    edge_alpha_max<<<(EP + 255) / 256, 256, 0, stream>>>(EI, E, EP, SS, SD, AEX, AMAX);
    edge_exp_sum<<<(EP + 255) / 256, 256, 0, stream>>>(EI, E, EP, AEX, AMAX, DEN);
    scatter_edges<<<EP, outd, 0, stream>>>(EI, E, H, AEX, DEN, ACC, outd);
    bias_elu<<<(total + 255) / 256, 256, 0, stream>>>(ACC, bv, xout, total, outd);
  };

  layer(X,  W1, as1, ad1, b1, 128, 128, XB);
  layer(XB, W2, as2, ad2, b2, 128,  64, XB);
  layer(XB, W3, as3, ad3, b3,  64,  32, XB);
  final_linear_sigmoid<<<(Nn + 7) / 8, 256, 0, stream>>>(XB, Wl, bl,
                                                         (float*)d_out, Nn);
}